// GlobalLinearSelfAttention_68212670595883
// MI455X (gfx1250) — compile-verified
//
#include <hip/hip_runtime.h>
#include <math.h>

#define HEADS 16
#define DHEAD 64
#define DIMX  1024
#define INNER 1024
#define BSZ   2
#define SEQ   2048
#define BH    (BSZ*HEADS)   // 32

typedef float v2f __attribute__((ext_vector_type(2)));
typedef float v8f __attribute__((ext_vector_type(8)));

// D = A(16x4 f32) * B(4x16 f32) + C(16x16 f32)   -> v_wmma_f32_16x16x4_f32
__device__ __forceinline__ v8f wmma4(v2f a, v2f b, v8f c) {
  return __builtin_amdgcn_wmma_f32_16x16x4_f32(false, a, false, b, (short)0, c,
                                               false, false);
}

// CDNA5 async global->LDS copy (ASYNCcnt-tracked), per-lane 16B.
__device__ __forceinline__ void async_g2l_b128(void* lds_ptr, const void* gptr) {
  unsigned loff = (unsigned)(unsigned long long)lds_ptr;  // low 32b = LDS offset
  asm volatile("global_load_async_to_lds_b128 %0, %1, off"
               :: "v"(loff), "v"(gptr) : "memory");
}
__device__ __forceinline__ void async_wait0() {
  asm volatile("s_wait_asynccnt 0x0" ::: "memory");
}

// ---------------- K0: rope table [2048 x 64] -------------------------------
__global__ void rope_kernel(float* __restrict__ rope) {
  int i = blockIdx.x * blockDim.x + threadIdx.x;   // 0..131071
  int n = i >> 6, d = i & 63;
  int j = d & 31;
  float invf = __powf(10000.0f, -(float)(2 * j) / 64.0f);
  float ang  = (float)n * invf;
  rope[i] = (d < 32) ? __sinf(ang) : __cosf(ang);
}

// ---------------- K1: qkv GEMM 4096x3072 (K=1024), 128x128 block tiles -----
// 512 threads = 16 waves; wave (wr,wc) owns 2x2 16x16 subtiles; K-chunk 32.
__global__ void qkv_kernel(const float* __restrict__ feats,
                           const float* __restrict__ Wqkv,
                           const float* __restrict__ rope,
                           const unsigned char* __restrict__ mask,
                           float* __restrict__ q, float* __restrict__ k,
                           float* __restrict__ v) {
  __shared__ __align__(16) float la[128 * 32];
  __shared__ __align__(16) float lb[32 * 128];
  const int t     = threadIdx.x;
  const int lane  = t & 31;
  const int wave  = t >> 5;                 // 0..15
  const int wr    = wave >> 2, wc = wave & 3;
  const int klane = (lane >> 4) * 2;
  const int row0  = (blockIdx.x / 24) * 128;   // 32 row-blocks
  const int col0  = (blockIdx.x % 24) * 128;   // 24 col-blocks
  v8f acc00 = {}, acc01 = {}, acc10 = {}, acc11 = {};
  const int m0 = wr * 32 + (lane & 15), m1 = m0 + 16;
  const int c0 = wc * 32 + (lane & 15), c1 = c0 + 16;

  for (int kb = 0; kb < DIMX; kb += 32) {
    __syncthreads();
    // A: 128x32 tile (1024 float4, 2 per thread); B: 32x128 tile likewise
    for (int u = 0; u < 2; ++u) {
      int f  = t + u * 512;
      int ar = f >> 3, ac = (f & 7) * 4;
      async_g2l_b128(&la[ar * 32 + ac],
                     &feats[(long)(row0 + ar) * DIMX + kb + ac]);
      int br = f >> 5, bc = (f & 31) * 4;
      async_g2l_b128(&lb[br * 128 + bc],
                     &Wqkv[(long)(kb + br) * 3072 + col0 + bc]);
    }
    async_wait0();
    __syncthreads();
#pragma unroll
    for (int kk = 0; kk < 32; kk += 4) {
      v2f a0, a1, b0, b1;
      a0.x = la[m0 * 32 + kk + klane];     a0.y = la[m0 * 32 + kk + klane + 1];
      a1.x = la[m1 * 32 + kk + klane];     a1.y = la[m1 * 32 + kk + klane + 1];
      b0.x = lb[(kk + klane) * 128 + c0];  b0.y = lb[(kk + klane + 1) * 128 + c0];
      b1.x = lb[(kk + klane) * 128 + c1];  b1.y = lb[(kk + klane + 1) * 128 + c1];
      acc00 = wmma4(a0, b0, acc00);
      acc01 = wmma4(a0, b1, acc01);
      acc10 = wmma4(a1, b0, acc10);
      acc11 = wmma4(a1, b1, acc11);
    }
  }
  // scatter epilogue: q/k get rope, v gets mask
  v8f accs[4] = {acc00, acc01, acc10, acc11};
  for (int sr = 0; sr < 2; ++sr)
    for (int sc = 0; sc < 2; ++sc) {
      v8f a = accs[sr * 2 + sc];
      int cc = col0 + wc * 32 + sc * 16 + (lane & 15);
      int which = cc / INNER;
      int h = (cc % INNER) >> 6;
      int d = cc & 63;
      for (int r = 0; r < 8; ++r) {
        int mm = row0 + wr * 32 + sr * 16 + r + 8 * (lane >> 4);
        int bi = mm >> 11, n = mm & 2047;
        long dst = ((long)(bi * HEADS + h) * SEQ + n) * DHEAD + d;
        float val = a[r];
        if (which == 0)      q[dst] = val + rope[n * DHEAD + d];
        else if (which == 1) k[dst] = val + rope[n * DHEAD + d];
        else                 v[dst] = mask[bi * SEQ + n] ? val : 0.0f;
      }
    }
}

// ---------------- K2: softmax over d for q, fused *scale -------------------
__global__ void qsoftmax_kernel(float* __restrict__ q) {
  const int lane = threadIdx.x & 31;
  const int wave = threadIdx.x >> 5;
  const long row = (long)blockIdx.x * 8 + wave;     // 0..65535
  float x0 = q[row * 64 + lane];
  float x1 = q[row * 64 + 32 + lane];
  float m = fmaxf(x0, x1);
  for (int off = 16; off > 0; off >>= 1) m = fmaxf(m, __shfl_xor(m, off, 32));
  float e0 = __expf(x0 - m), e1 = __expf(x1 - m);
  float s = e0 + e1;
  for (int off = 16; off > 0; off >>= 1) s += __shfl_xor(s, off, 32);
  float inv = 0.125f / s;                            // scale = d^-0.5 = 1/8
  q[row * 64 + lane]      = e0 * inv;
  q[row * 64 + 32 + lane] = e1 * inv;
}

// ---------------- K3: softmax over n for k (masked), block per (b,h) -------
__global__ void ksoftmax_kernel(float* __restrict__ k,
                                const unsigned char* __restrict__ mask) {
  __shared__ float red[16 * 64];
  __shared__ float rowred[64];
  const int bh  = blockIdx.x;
  const int b   = bh >> 4;
  const int d   = threadIdx.x & 63;
  const int seg = threadIdx.x >> 6;                 // 0..15
  const int n0  = seg * 128;
  const float NEG = -3.402823466e38f;
  const float* kbp = k + (long)bh * SEQ * 64;

  float lmax = NEG;
  for (int n = n0; n < n0 + 128; ++n) {
    float val = mask[b * SEQ + n] ? kbp[(long)n * 64 + d] : NEG;
    lmax = fmaxf(lmax, val);
  }
  red[seg * 64 + d] = lmax;
  __syncthreads();
  if (seg == 0) {
    float mx = red[d];
    for (int s2 = 1; s2 < 16; ++s2) mx = fmaxf(mx, red[s2 * 64 + d]);
    rowred[d] = mx;
  }
  __syncthreads();
  const float mx = rowred[d];
  float lsum = 0.f;
  for (int n = n0; n < n0 + 128; ++n) {
    float val = mask[b * SEQ + n] ? kbp[(long)n * 64 + d] : NEG;
    lsum += __expf(val - mx);
  }
  red[seg * 64 + d] = lsum;
  __syncthreads();
  if (seg == 0) {
    float s = 0.f;
    for (int s2 = 0; s2 < 16; ++s2) s += red[s2 * 64 + d];
    rowred[d] = s;
  }
  __syncthreads();
  const float inv = 1.0f / rowred[d];
  float* kw = k + (long)bh * SEQ * 64;
  for (int n = n0; n < n0 + 128; ++n) {
    long idx = (long)n * 64 + d;
    float val = mask[b * SEQ + n] ? kw[idx] : NEG;
    kw[idx] = __expf(val - mx) * inv;
  }
}

// ---------------- K4: ctx = K^T V (64x64, K=2048) via async LDS chunks -----
__global__ void context_kernel(const float* __restrict__ k,
                               const float* __restrict__ v,
                               float* __restrict__ ctx) {
  __shared__ __align__(16) float lk[64 * 64];
  __shared__ __align__(16) float lv[64 * 64];
  const int bh    = blockIdx.x;
  const int lane  = threadIdx.x & 31;
  const int wave  = threadIdx.x >> 5;   // 0..15
  const int td    = wave >> 2, te = wave & 3;
  const int klane = (lane >> 4) * 2;
  const float* kbp = k + (long)bh * SEQ * 64;
  const float* vbp = v + (long)bh * SEQ * 64;
  v8f acc = {};
  for (int c = 0; c < SEQ; c += 64) {
    __syncthreads();
    for (int j = 0; j < 8; j += 4) {
      int idx = threadIdx.x * 8 + j;
      async_g2l_b128(&lk[idx], &kbp[(long)c * 64 + idx]);
      async_g2l_b128(&lv[idx], &vbp[(long)c * 64 + idx]);
    }
    async_wait0();
    __syncthreads();
#pragma unroll
    for (int kk = 0; kk < 64; kk += 4) {
      v2f a, b;
      a.x = lk[(kk + klane) * 64 + td * 16 + (lane & 15)];      // A[d][n]=k[n][d]
      a.y = lk[(kk + klane + 1) * 64 + td * 16 + (lane & 15)];
      b.x = lv[(kk + klane) * 64 + te * 16 + (lane & 15)];      // B[n][e]=v[n][e]
      b.y = lv[(kk + klane + 1) * 64 + te * 16 + (lane & 15)];
      acc = wmma4(a, b, acc);
    }
  }
  for (int r = 0; r < 8; ++r) {
    int dd = td * 16 + r + 8 * (lane >> 4);
    int ee = te * 16 + (lane & 15);
    ctx[(long)bh * 4096 + dd * 64 + ee] = acc[r];
  }
}

// ---------------- K5: oi = Q @ ctx -> [b,n,h*64] layout --------------------
__global__ void qctx_kernel(const float* __restrict__ q,
                            const float* __restrict__ ctx,
                            float* __restrict__ oi) {
  const int lane  = threadIdx.x & 31;
  const int wave  = threadIdx.x >> 5;
  const int tile  = blockIdx.x * 8 + wave;   // 0..16383
  const int bh    = tile >> 9;
  const int rem   = tile & 511;
  const int tn    = rem >> 2;                // 0..127
  const int te    = rem & 3;
  const int klane = (lane >> 4) * 2;
  const float* qb = q + (long)bh * SEQ * 64;
  const float* cb = ctx + (long)bh * 4096;
  const int mrow  = tn * 16 + (lane & 15);
  const int ecol  = te * 16 + (lane & 15);
  v8f acc = {};
#pragma unroll
  for (int kb = 0; kb < 64; kb += 4) {
    v2f a, b;
    a.x = qb[(long)mrow * 64 + kb + klane];
    a.y = qb[(long)mrow * 64 + kb + klane + 1];
    b.x = cb[(kb + klane) * 64 + ecol];
    b.y = cb[(kb + klane + 1) * 64 + ecol];
    acc = wmma4(a, b, acc);
  }
  const int bi = bh >> 4, h = bh & 15;
  for (int r = 0; r < 8; ++r) {
    int n = tn * 16 + r + 8 * (lane >> 4);
    oi[((long)bi * SEQ + n) * INNER + h * DHEAD + ecol] = acc[r];
  }
}

// ---------------- K6: out = oi @ W_out + b_out, 128x128 block tiles --------
__global__ void outproj_kernel(const float* __restrict__ oi,
                               const float* __restrict__ Wout,
                               const float* __restrict__ bout,
                               float* __restrict__ out) {
  __shared__ __align__(16) float la[128 * 32];
  __shared__ __align__(16) float lb[32 * 128];
  const int t     = threadIdx.x;
  const int lane  = t & 31;
  const int wave  = t >> 5;
  const int wr    = wave >> 2, wc = wave & 3;
  const int klane = (lane >> 4) * 2;
  const int row0  = (blockIdx.x >> 3) * 128;   // 32 row-blocks
  const int col0  = (blockIdx.x & 7) * 128;    // 8 col-blocks
  v8f acc00 = {}, acc01 = {}, acc10 = {}, acc11 = {};
  const int m0 = wr * 32 + (lane & 15), m1 = m0 + 16;
  const int c0 = wc * 32 + (lane & 15), c1 = c0 + 16;

  for (int kb = 0; kb < INNER; kb += 32) {
    __syncthreads();
    for (int u = 0; u < 2; ++u) {
      int f  = t + u * 512;
      int ar = f >> 3, ac = (f & 7) * 4;
      async_g2l_b128(&la[ar * 32 + ac],
                     &oi[(long)(row0 + ar) * INNER + kb + ac]);
      int br = f >> 5, bc = (f & 31) * 4;
      async_g2l_b128(&lb[br * 128 + bc],
                     &Wout[(long)(kb + br) * DIMX + col0 + bc]);
    }
    async_wait0();
    __syncthreads();
#pragma unroll
    for (int kk = 0; kk < 32; kk += 4) {
      v2f a0, a1, b0, b1;
      a0.x = la[m0 * 32 + kk + klane];     a0.y = la[m0 * 32 + kk + klane + 1];
      a1.x = la[m1 * 32 + kk + klane];     a1.y = la[m1 * 32 + kk + klane + 1];
      b0.x = lb[(kk + klane) * 128 + c0];  b0.y = lb[(kk + klane + 1) * 128 + c0];
      b1.x = lb[(kk + klane) * 128 + c1];  b1.y = lb[(kk + klane + 1) * 128 + c1];
      acc00 = wmma4(a0, b0, acc00);
      acc01 = wmma4(a0, b1, acc01);
      acc10 = wmma4(a1, b0, acc10);
      acc11 = wmma4(a1, b1, acc11);
    }
  }
  v8f accs[4] = {acc00, acc01, acc10, acc11};
  for (int sr = 0; sr < 2; ++sr)
    for (int sc = 0; sc < 2; ++sc) {
      v8f a = accs[sr * 2 + sc];
      int cc = col0 + wc * 32 + sc * 16 + (lane & 15);
      float bo = bout[cc];
      for (int r = 0; r < 8; ++r) {
        int mm = row0 + wr * 32 + sr * 16 + r + 8 * (lane >> 4);
        out[(long)mm * DIMX + cc] = a[r] + bo;
      }
    }
}

// ---------------- K7: attn = Q K^T * scale ---------------------------------
// Block = 8 waves sharing one k-tile staged via async LDS; each wave keeps its
// full Q 16x64 A-operand in registers for all 128 m-tiles. NT stores (537 MB).
__global__ void attn_kernel(const float* __restrict__ q,
                            const float* __restrict__ k,
                            float* __restrict__ attn) {
  __shared__ __align__(16) float lkt[16 * 64];    // one 16x64 k tile (4KB)
  const int t     = threadIdx.x;                  // 256
  const int lane  = t & 31;
  const int wave  = t >> 5;                       // 0..7
  const int bh    = blockIdx.x >> 4;              // 32
  const int ng    = blockIdx.x & 15;              // 16 n-groups
  const int tn    = ng * 8 + wave;                // 0..127
  const int klane = (lane >> 4) * 2;
  const float* qb  = q + (long)bh * SEQ * 64;
  const float* kbp = k + (long)bh * SEQ * 64;
  const int nrow   = tn * 16 + (lane & 15);

  // preload this wave's A operand (16x64 q tile) into registers
  v2f areg[16];
#pragma unroll
  for (int s = 0; s < 16; ++s) {
    areg[s].x = qb[(long)nrow * 64 + s * 4 + klane];
    areg[s].y = qb[(long)nrow * 64 + s * 4 + klane + 1];
  }

  float* ab = attn + (long)bh * SEQ * SEQ;
  for (int tm = 0; tm < 128; ++tm) {
    __syncthreads();
    async_g2l_b128(&lkt[t * 4], &kbp[(long)tm * 1024 + t * 4]);
    async_wait0();
    __syncthreads();
    v8f acc = {};
#pragma unroll
    for (int s = 0; s < 16; ++s) {
      v2f b;                                       // B[d][m] = k[m][d]
      b.x = lkt[(lane & 15) * 64 + s * 4 + klane];
      b.y = lkt[(lane & 15) * 64 + s * 4 + klane + 1];
      acc = wmma4(areg[s], b, acc);
    }
    for (int r = 0; r < 8; ++r) {
      int n = tn * 16 + r + 8 * (lane >> 4);
      __builtin_nontemporal_store(acc[r] * 0.125f,
                                  &ab[(long)n * SEQ + tm * 16 + (lane & 15)]);
    }
  }
}

extern "C" void kernel_launch(void* const* d_in, const int* in_sizes, int n_in,
                              void* d_out, int out_size, void* d_ws,
                              size_t ws_size, hipStream_t stream) {
  (void)in_sizes; (void)n_in; (void)out_size; (void)ws_size;
  const float* feats = (const float*)d_in[0];
  const float* Wqkv  = (const float*)d_in[1];
  const float* Wout  = (const float*)d_in[2];
  const float* bout  = (const float*)d_in[3];
  const unsigned char* mask = (const unsigned char*)d_in[4];

  float* out  = (float*)d_out;                        // [2,2048,1024]
  float* attn = out + (long)BSZ * SEQ * DIMX;         // [2,16,2048,2048]

  float* ws   = (float*)d_ws;
  float* q    = ws;                                   // 4,194,304 f32
  float* k    = q + (long)BH * SEQ * DHEAD;
  float* v    = k + (long)BH * SEQ * DHEAD;
  float* rope = v + (long)BH * SEQ * DHEAD;           // 131,072
  float* ctx  = rope + SEQ * DHEAD;                   // 131,072
  float* oi   = ctx + BH * DHEAD * DHEAD;             // 4,194,304

  rope_kernel    <<<512,  256, 0, stream>>>(rope);
  qkv_kernel     <<<768,  512, 0, stream>>>(feats, Wqkv, rope, mask, q, k, v);
  qsoftmax_kernel<<<8192, 256, 0, stream>>>(q);
  ksoftmax_kernel<<<32,  1024, 0, stream>>>(k, mask);
  context_kernel <<<32,   512, 0, stream>>>(k, v, ctx);
  qctx_kernel    <<<2048, 256, 0, stream>>>(q, ctx, oi);
  outproj_kernel <<<256,  512, 0, stream>>>(oi, Wout, bout, out);
  attn_kernel    <<<512,  256, 0, stream>>>(q, k, attn);
}